// GraphSAGE_79164837200035
// MI455X (gfx1250) — compile-verified
//
#include <hip/hip_runtime.h>

typedef float v2f __attribute__((ext_vector_type(2)));
typedef float v8f __attribute__((ext_vector_type(8)));

#define D 64

// -------- degree: one float atomic per edge (deg fits in L2, resolved there) ----
__global__ __launch_bounds__(256) void deg_kernel(const int* __restrict__ dst,
                                                  float* __restrict__ deg, int E) {
    int e = blockIdx.x * blockDim.x + threadIdx.x;
    if (e < E) unsafeAtomicAdd(&deg[dst[e]], 1.0f);
}

// -------- scatter-sum: one wave32 per edge; lane handles features d and d+32 ----
// Gather from X is two coalesced 128B transactions per edge; scatter uses
// hardware global_atomic_add_f32 hitting L2 (msg buffer is 25.6MB << 192MB L2).
__global__ __launch_bounds__(256) void scatter_kernel(const float* __restrict__ X,
                                                      const int* __restrict__ src,
                                                      const int* __restrict__ dst,
                                                      float* __restrict__ msg, int E) {
    int gid  = blockIdx.x * blockDim.x + threadIdx.x;
    int e    = gid >> 5;
    int lane = gid & 31;
    if (e >= E) return;
    int s = src[e];
    int d = dst[e];
    float v0 = X[(size_t)s * D + lane];
    float v1 = X[(size_t)s * D + lane + 32];
    unsafeAtomicAdd(&msg[(size_t)d * D + lane], v0);
    unsafeAtomicAdd(&msg[(size_t)d * D + lane + 32], v1);
}

// -------- fused SAGE layer: out = act( X@Ws^T + (msg/deg)@Wn^T + b ) ------------
// 4 waves/block; block = 16-node row tile (LDS-staged), wave = 16-col output tile.
// 32x v_wmma_f32_16x16x4_f32 per wave (16 K-steps x {self, neigh}).
template <bool RELU>
__global__ __launch_bounds__(128) void sage_layer_wmma(
    const float* __restrict__ X, const float* __restrict__ msg,
    const float* __restrict__ deg,
    const float* __restrict__ Wself, const float* __restrict__ Wneigh,
    const float* __restrict__ bias, float* __restrict__ out, int n) {

    __shared__ float xs[16][D + 1];   // +1 pad: stride 65 -> conflict-free column reads
    __shared__ float hn[16][D + 1];

    const int tid     = threadIdx.x;
    const int rowbase = blockIdx.x * 16;

    __builtin_prefetch(Wself, 0, 0);   // global_prefetch_b8 (weights are 16KB each)
    __builtin_prefetch(Wneigh, 0, 0);

    // Cooperative fill: 2 x (16x64) tiles, float4 coalesced; fuse mean-divide.
    for (int i = tid; i < 256; i += 128) {
        int flat = i * 4;
        int r    = flat >> 6;
        int c    = flat & (D - 1);
        int row  = rowbase + r;
        float4 xv = make_float4(0.f, 0.f, 0.f, 0.f);
        float4 mv = make_float4(0.f, 0.f, 0.f, 0.f);
        float invd = 1.0f;
        if (row < n) {
            xv   = *(const float4*)(X   + (size_t)row * D + c);
            mv   = *(const float4*)(msg + (size_t)row * D + c);
            invd = 1.0f / fmaxf(deg[row], 1.0f);
        }
        xs[r][c + 0] = xv.x; xs[r][c + 1] = xv.y;
        xs[r][c + 2] = xv.z; xs[r][c + 3] = xv.w;
        hn[r][c + 0] = mv.x * invd; hn[r][c + 1] = mv.y * invd;
        hn[r][c + 2] = mv.z * invd; hn[r][c + 3] = mv.w * invd;
    }
    __syncthreads();

    const int lane  = tid & 31;
    const int wv    = tid >> 5;      // wave id 0..3 -> output column tile
    const int obase = wv * 16;
    const int kh    = lane >> 4;     // K-half: lanes 0-15 -> K+0,1 ; 16-31 -> K+2,3
    const int mr    = lane & 15;     // A row / B column within tile
    const int ocol  = obase + mr;

    v8f c_acc = {};
    for (int kb = 0; kb < D; kb += 4) {
        const int k0 = kb + kh * 2;
        v2f a_s, a_n, b_s, b_n;
        // A 16x4 f32 layout (ISA 7.12.2): M = lane&15, VGPR0/1 = K pair per half
        a_s[0] = xs[mr][k0];  a_s[1] = xs[mr][k0 + 1];
        a_n[0] = hn[mr][k0];  a_n[1] = hn[mr][k0 + 1];
        // B 4x16: row K striped across lane halves; B[k][o] = W[o][k]
        b_s[0] = Wself[ocol * D + k0];   b_s[1] = Wself[ocol * D + k0 + 1];
        b_n[0] = Wneigh[ocol * D + k0];  b_n[1] = Wneigh[ocol * D + k0 + 1];
        c_acc = __builtin_amdgcn_wmma_f32_16x16x4_f32(
            false, a_s, false, b_s, (short)0, c_acc, false, false);
        c_acc = __builtin_amdgcn_wmma_f32_16x16x4_f32(
            false, a_n, false, b_n, (short)0, c_acc, false, false);
    }

    // C/D 16x16 f32 layout: lane half selects M base (0 or 8), VGPR v = M offset
    const float bv   = bias[ocol];
    const int   row0 = rowbase + kh * 8;
#pragma unroll
    for (int v = 0; v < 8; ++v) {
        int row = row0 + v;
        if (row < n) {
            float val = c_acc[v] + bv;
            if (RELU) val = fmaxf(val, 0.0f);
            out[(size_t)row * D + ocol] = val;
        }
    }
}

extern "C" void kernel_launch(void* const* d_in, const int* in_sizes, int n_in,
                              void* d_out, int out_size, void* d_ws, size_t ws_size,
                              hipStream_t stream) {
    const float* in_feat = (const float*)d_in[0];
    const int*   src     = (const int*)d_in[1];
    const int*   dst     = (const int*)d_in[2];
    const float* Ws1     = (const float*)d_in[3];
    const float* Wn1     = (const float*)d_in[4];
    const float* b1      = (const float*)d_in[5];
    const float* Ws2     = (const float*)d_in[6];
    const float* Wn2     = (const float*)d_in[7];
    const float* b2      = (const float*)d_in[8];
    float*       out     = (float*)d_out;

    const int n = in_sizes[0] / D;     // 100000
    const int E = in_sizes[1];         // 1600000

    // Workspace layout (floats): [deg n][pad][msg n*D][h1 n*D]  ~51.6 MB
    float* ws      = (float*)d_ws;
    size_t msgOff  = ((size_t)n + 255) & ~(size_t)255;
    size_t h1Off   = msgOff + (size_t)n * D;
    float* deg     = ws;
    float* msg     = ws + msgOff;
    float* h1      = ws + h1Off;

    const int degBlocks     = (E + 255) / 256;
    const int scatterBlocks = (E * 32 + 255) / 256;   // one wave per edge
    const int layerBlocks   = (n + 15) / 16;

    // zero deg + msg (graph-capturable stream memset)
    hipMemsetAsync(ws, 0, (msgOff + (size_t)n * D) * sizeof(float), stream);

    // degree once (same graph in both layers)
    deg_kernel<<<degBlocks, 256, 0, stream>>>(dst, deg, E);

    // layer 1
    scatter_kernel<<<scatterBlocks, 256, 0, stream>>>(in_feat, src, dst, msg, E);
    sage_layer_wmma<true><<<layerBlocks, 128, 0, stream>>>(
        in_feat, msg, deg, Ws1, Wn1, b1, h1, n);

    // layer 2
    hipMemsetAsync(msg, 0, (size_t)n * D * sizeof(float), stream);
    scatter_kernel<<<scatterBlocks, 256, 0, stream>>>(h1, src, dst, msg, E);
    sage_layer_wmma<false><<<layerBlocks, 128, 0, stream>>>(
        h1, msg, deg, Ws2, Wn2, b2, out, n);
}